// ModelVLLM_28681791602756
// MI455X (gfx1250) — compile-verified
//
#include <hip/hip_runtime.h>

// Lightning-attention linear decode step, fp32, memory-bound (~1 FLOP/byte).
// One wave32 per (b,h): streams the 16KB KV tile once, accumulates
// q @ KV via V_WMMA_F32_16X16X4_F32 with q broadcast across the M dimension,
// then applies out = decay * (q@KV) + (q.k) * v.

typedef __attribute__((ext_vector_type(2))) float v2f;
typedef __attribute__((ext_vector_type(8))) float v8f;

#define B_DIM 256
#define H_DIM 64
#define D_DIM 64

__global__ __launch_bounds__(256) void lightning_decode_wmma(
    const float* __restrict__ q,
    const float* __restrict__ k,
    const float* __restrict__ v,
    const float* __restrict__ kv_caches,
    const float* __restrict__ slope_rate,
    const long long* __restrict__ slot_idx,
    float* __restrict__ out)
{
    const int wave = threadIdx.x >> 5;          // 0..7
    const int lane = threadIdx.x & 31;          // 0..31
    const int b    = blockIdx.x >> 3;           // 0..255
    const int h    = ((blockIdx.x & 7) << 3) | wave;  // 0..63

    const size_t bh = (size_t)(b * H_DIM + h);
    const float* qp = q + bh * D_DIM;
    const float* kp = k + bh * D_DIM;
    const float* vp = v + bh * D_DIM;
    const long long slot = slot_idx[b];
    const float* kvp = kv_caches + ((size_t)slot * H_DIM + h) * (size_t)(D_DIM * D_DIM);

    // ---- q . k  (rank-1 epilogue coefficient), wave32 reduction ----
    float qk = qp[lane] * kp[lane] + qp[lane + 32] * kp[lane + 32];
    #pragma unroll
    for (int off = 16; off >= 1; off >>= 1)
        qk += __shfl_xor(qk, off, 32);

    const int half = lane >> 4;     // 0: lanes 0-15 (K=0,1) | 1: lanes 16-31 (K=2,3)
    const int col  = lane & 15;     // N within a 16-wide tile

    // Four 16-wide output tiles covering e = 0..63
    v8f acc0 = {}, acc1 = {}, acc2 = {}, acc3 = {};

    // ---- q @ KV : 16 K-steps of V_WMMA_F32_16X16X4_F32 per N-tile ----
    #pragma unroll 4
    for (int k0 = 0; k0 < D_DIM; k0 += 4) {
        const int ra = k0 + half * 2;   // base row this lane-group supplies

        // A operand (16x4 f32): q[ra], q[ra+1] broadcast to all 16 M rows.
        // ISA layout: VGPR0 = K=0 (lanes 0-15) / K=2 (lanes 16-31); VGPR1 = K=1 / K=3.
        v2f a = *(const v2f*)(qp + ra);

        // B operand (4x16 f32): VGPR0 = row ra, VGPR1 = row ra+1, N = col.
        // Each 16-lane half reads 64 contiguous bytes per load (coalesced);
        // every KV element is touched exactly once.
        const float* brow = kvp + (size_t)ra * D_DIM + col;
        v2f b0, b1, b2, b3;
        b0.x = brow[0];   b0.y = brow[D_DIM + 0];
        b1.x = brow[16];  b1.y = brow[D_DIM + 16];
        b2.x = brow[32];  b2.y = brow[D_DIM + 32];
        b3.x = brow[48];  b3.y = brow[D_DIM + 48];

        acc0 = __builtin_amdgcn_wmma_f32_16x16x4_f32(false, a, false, b0, (short)0, acc0, false, false);
        acc1 = __builtin_amdgcn_wmma_f32_16x16x4_f32(false, a, false, b1, (short)0, acc1, false, false);
        acc2 = __builtin_amdgcn_wmma_f32_16x16x4_f32(false, a, false, b2, (short)0, acc2, false, false);
        acc3 = __builtin_amdgcn_wmma_f32_16x16x4_f32(false, a, false, b3, (short)0, acc3, false, false);
    }

    // ---- epilogue: out[e] = decay * (q@KV)[e] + qk * v[e] ----
    const float decay = __expf(-slope_rate[h]);
    float* op = out + bh * D_DIM;

    // D rows are replicated (A rows identical), so element 0 of each
    // accumulator holds the value for N=col (M=0 for lanes 0-15, M=8 for 16-31).
    float r0, r1; int e0, e1;
    if (half == 0) { r0 = acc0[0]; e0 = col;      r1 = acc1[0]; e1 = col + 16; }
    else           { r0 = acc2[0]; e0 = col + 32; r1 = acc3[0]; e1 = col + 48; }

    op[e0] = decay * r0 + qk * vp[e0];
    op[e1] = decay * r1 + qk * vp[e1];
}

extern "C" void kernel_launch(void* const* d_in, const int* in_sizes, int n_in,
                              void* d_out, int out_size, void* d_ws, size_t ws_size,
                              hipStream_t stream) {
    const float*     q     = (const float*)d_in[0];
    const float*     k     = (const float*)d_in[1];
    const float*     v     = (const float*)d_in[2];
    const float*     kv    = (const float*)d_in[3];
    const float*     slope = (const float*)d_in[4];
    const long long* slot  = (const long long*)d_in[5];   // int64 per reference
    float*           out   = (float*)d_out;

    dim3 grid(B_DIM * (H_DIM / 8));   // 2048 blocks
    dim3 block(256);                  // 8 waves -> 8 heads of one batch element
    hipLaunchKernelGGL(lightning_decode_wmma, grid, block, 0, stream,
                       q, k, v, kv, slope, slot, out);
}